// UOTloss_49615462203960
// MI455X (gfx1250) — compile-verified
//
#include <hip/hip_runtime.h>
#include <hip/hip_bf16.h>
#include <math.h>

// UOT mm_unbalanced (div='kl', reg_m=0.1, 30 iters) for m=n=4096, d=256.
// Strategy:
//   s = x@y^T via V_WMMA_F32_16X16X4_F32 (fp32 matrix op, 8.6 GFLOP, tiny)
//   K = exp(-5*M)/4096 with M = xx_i + yy_j - 2 s   (epilogue, v_exp_f32)
//   G1 = K/4096 (first MM step is closed form since G0 = a b^T is constant)
//   29 x { update G <- K*G/(sqrt(R_i*C_j)+eps), fused new-rowsum; colsum pass }
//   out = dot(colsum(G30), colsum(M)) / m^2,  colsum(M) computed analytically.
// K+G = 128MB -> resident in the 192MB L2; ~7.5GB of L2 traffic total.
// No atomics -> deterministic. Workspace need: ~129 MB.

#define UOT_M 4096
#define UOT_N 4096
#define UOT_D 256
#define UOT_EPS 1e-16f

typedef float v2f_t __attribute__((ext_vector_type(2)));
typedef float v8f_t __attribute__((ext_vector_type(8)));

__device__ __forceinline__ float uot_block_reduce(float v, float* sdata) {
  int tid = threadIdx.x;
  sdata[tid] = v;
  __syncthreads();
  for (int s = 128; s > 0; s >>= 1) {
    if (tid < s) sdata[tid] += sdata[tid + s];
    __syncthreads();
  }
  return sdata[0];
}

// per-row squared norm: out[row] = sum_d v[row,d]^2   (grid 16 x 256)
__global__ void uot_rownorm2(const float* __restrict__ v, float* __restrict__ out) {
  int row = blockIdx.x * blockDim.x + threadIdx.x;
  const float* p = v + (size_t)row * UOT_D;
  float acc = 0.f;
  for (int d = 0; d < UOT_D; ++d) { float t = p[d]; acc += t * t; }
  out[row] = acc;
}

// column sums of x (stage 1): block b sums rows [b*128, b*128+128)  (grid 32 x 256)
__global__ void uot_xsum1(const float* __restrict__ x, float* __restrict__ part) {
  int d = threadIdx.x;
  int b = blockIdx.x;
  const float* p = x + (size_t)b * 128 * UOT_D + d;
  float acc = 0.f;
  for (int i = 0; i < 128; ++i) acc += p[(size_t)i * UOT_D];
  part[b * UOT_D + d] = acc;
}
__global__ void uot_xsum2(const float* __restrict__ part, float* __restrict__ xsum) {
  int d = threadIdx.x;
  float acc = 0.f;
  for (int b = 0; b < 32; ++b) acc += part[b * UOT_D + d];
  xsum[d] = acc;
}

// scalar sum of a length-n vector (1 x 256)
__global__ void uot_vecsum(const float* __restrict__ v, int n, float* __restrict__ out) {
  __shared__ float sdata[256];
  float acc = 0.f;
  for (int i = threadIdx.x; i < n; i += 256) acc += v[i];
  float r = uot_block_reduce(acc, sdata);
  if (threadIdx.x == 0) out[0] = r;
}

// colM[j] = sum_i M[i,j] = sxx + m*yy[j] - 2 * dot(xsum, y[j,:])   (grid 16 x 256)
__global__ void uot_colm(const float* __restrict__ y, const float* __restrict__ xsum,
                         const float* __restrict__ yy, const float* __restrict__ sxx,
                         float* __restrict__ colM) {
  __shared__ float xs[UOT_D];
  for (int d = threadIdx.x; d < UOT_D; d += blockDim.x) xs[d] = xsum[d];
  __syncthreads();
  int j = blockIdx.x * blockDim.x + threadIdx.x;
  const float* yr = y + (size_t)j * UOT_D;
  float acc = 0.f;
  for (int d = 0; d < UOT_D; ++d) acc += xs[d] * yr[d];
  colM[j] = sxx[0] + (float)UOT_M * yy[j] - 2.0f * acc;
}

// WMMA fp32 GEMM: s = x @ y^T; epilogue writes K = exp(-5M)/4096 and G = K/4096.
// Block = 256 thr = 8 waves; wave w owns rows [by*128+w*16, +16), cols [bx*64, +64).
// A 16x4 frag : lane holds float2 x[row0+(lane&15), k0+2*(lane>>4) .. +1]
// B 4x16 frag : lane holds float2 y[col0+(lane&15), k0+2*(lane>>4) .. +1]
// C/D element : acc[t][v] -> row = row0+v+8*(lane>>4), col = col0+t*16+(lane&15)
__global__ __launch_bounds__(256) void uot_gemm_k(const float* __restrict__ x,
                                                  const float* __restrict__ y,
                                                  const float* __restrict__ xx,
                                                  const float* __restrict__ yy,
                                                  float* __restrict__ Kmat,
                                                  float* __restrict__ G) {
  const int wave = threadIdx.x >> 5;
  const int lane = threadIdx.x & 31;
  const int lo = lane & 15;
  const int hi = lane >> 4;
  const int row0 = blockIdx.y * 128 + wave * 16;
  const int col0 = blockIdx.x * 64;

  const float* ax = x + (size_t)(row0 + lo) * UOT_D + 2 * hi;
  const float* by = y + (size_t)(col0 + lo) * UOT_D + 2 * hi;

  v8f_t acc[4] = {};
#pragma unroll 4
  for (int k = 0; k < UOT_D; k += 4) {
    v2f_t a  = *(const v2f_t*)(ax + k);
    v2f_t b0 = *(const v2f_t*)(by + k);
    v2f_t b1 = *(const v2f_t*)(by + k + 16 * UOT_D);
    v2f_t b2 = *(const v2f_t*)(by + k + 32 * UOT_D);
    v2f_t b3 = *(const v2f_t*)(by + k + 48 * UOT_D);
    acc[0] = __builtin_amdgcn_wmma_f32_16x16x4_f32(false, a, false, b0, (short)0, acc[0], false, false);
    acc[1] = __builtin_amdgcn_wmma_f32_16x16x4_f32(false, a, false, b1, (short)0, acc[1], false, false);
    acc[2] = __builtin_amdgcn_wmma_f32_16x16x4_f32(false, a, false, b2, (short)0, acc[2], false, false);
    acc[3] = __builtin_amdgcn_wmma_f32_16x16x4_f32(false, a, false, b3, (short)0, acc[3], false, false);
  }

  const float inv4096 = 1.0f / 4096.0f;
#pragma unroll
  for (int t = 0; t < 4; ++t) {
    int col = col0 + t * 16 + lo;
    float yyv = yy[col];
#pragma unroll
    for (int vv = 0; vv < 8; ++vv) {
      int row = row0 + vv + 8 * hi;
      float s = acc[t][vv];
      float Mij = xx[row] + yyv - 2.0f * s;
      float kv = inv4096 * __expf(-5.0f * Mij);   // exp(-M/(2*reg_m)), reg_m=0.1
      size_t idx = (size_t)row * UOT_N + col;
      Kmat[idx] = kv;
      G[idx] = kv * inv4096;                      // closed-form first MM step
    }
  }
}

// initial row sums of G (grid 4096 x 256)
__global__ __launch_bounds__(256) void uot_rowsum(const float* __restrict__ G,
                                                  float* __restrict__ R) {
  __shared__ float sdata[256];
  const float4* g = (const float4*)(G + (size_t)blockIdx.x * UOT_N);
  float acc = 0.f;
  for (int j = threadIdx.x; j < UOT_N / 4; j += 256) {
    float4 v = g[j];
    acc += (v.x + v.y) + (v.z + v.w);
  }
  float r = uot_block_reduce(acc, sdata);
  if (threadIdx.x == 0) R[blockIdx.x] = r;
}

// column sums of G, 2 stages, no atomics (grid (16,8) x 256 ; then 16 x 256)
__global__ __launch_bounds__(256) void uot_colsum1(const float* __restrict__ G,
                                                   float* __restrict__ Cpart) {
  int j = blockIdx.x * 256 + threadIdx.x;
  const float* g = G + (size_t)blockIdx.y * 512 * UOT_N + j;
  float acc = 0.f;
  for (int i = 0; i < 512; ++i) acc += g[(size_t)i * UOT_N];
  Cpart[blockIdx.y * UOT_N + j] = acc;
}
__global__ void uot_colsum2(const float* __restrict__ Cpart, float* __restrict__ C) {
  int j = blockIdx.x * 256 + threadIdx.x;
  float acc = 0.f;
  for (int b = 0; b < 8; ++b) acc += Cpart[b * UOT_N + j];
  C[j] = acc;
}

// G <- K*G / (sqrt(R_i*C_j)+eps); fused new row-sum -> R[row]. (grid 4096 x 256)
__global__ __launch_bounds__(256) void uot_update(const float* __restrict__ Kmat,
                                                  float* __restrict__ G,
                                                  float* __restrict__ R,
                                                  const float* __restrict__ C) {
  __shared__ float sdata[256];
  int row = blockIdx.x;
  float r = R[row];
  const float4* kp = (const float4*)(Kmat + (size_t)row * UOT_N);
  float4* gp = (float4*)(G + (size_t)row * UOT_N);
  const float4* cp = (const float4*)C;
  float acc = 0.f;
  for (int j = threadIdx.x; j < UOT_N / 4; j += 256) {
    float4 kv = kp[j];
    float4 gv = gp[j];
    float4 cv = cp[j];
    gv.x = kv.x * gv.x / (sqrtf(r * cv.x) + UOT_EPS);
    gv.y = kv.y * gv.y / (sqrtf(r * cv.y) + UOT_EPS);
    gv.z = kv.z * gv.z / (sqrtf(r * cv.z) + UOT_EPS);
    gv.w = kv.w * gv.w / (sqrtf(r * cv.w) + UOT_EPS);
    gp[j] = gv;
    acc += (gv.x + gv.y) + (gv.z + gv.w);
  }
  float nr = uot_block_reduce(acc, sdata);
  if (threadIdx.x == 0) R[row] = nr;
}

// out = dot(C, colM) / m^2   (1 x 256)
__global__ void uot_final(const float* __restrict__ C, const float* __restrict__ colM,
                          float* __restrict__ out) {
  __shared__ float sdata[256];
  float acc = 0.f;
  for (int j = threadIdx.x; j < UOT_N; j += 256) acc += C[j] * colM[j];
  float r = uot_block_reduce(acc, sdata);
  if (threadIdx.x == 0) out[0] = r / ((float)UOT_M * (float)UOT_M);
}

extern "C" void kernel_launch(void* const* d_in, const int* in_sizes, int n_in,
                              void* d_out, int out_size, void* d_ws, size_t ws_size,
                              hipStream_t stream) {
  (void)in_sizes; (void)n_in; (void)out_size; (void)ws_size;
  const float* x = (const float*)d_in[0];
  const float* y = (const float*)d_in[1];
  float* out = (float*)d_out;

  // workspace layout (floats)
  float* Kmat  = (float*)d_ws;                    // 4096*4096
  float* G     = Kmat + (size_t)UOT_M * UOT_N;    // 4096*4096
  float* R     = G + (size_t)UOT_M * UOT_N;       // 4096
  float* C     = R + UOT_M;                       // 4096
  float* colM  = C + UOT_N;                       // 4096
  float* xxv   = colM + UOT_N;                    // 4096
  float* yyv   = xxv + UOT_M;                     // 4096
  float* Cpart = yyv + UOT_N;                     // 8*4096
  float* xpart = Cpart + 8 * UOT_N;               // 32*256
  float* xsum  = xpart + 32 * UOT_D;              // 256
  float* sxx   = xsum + UOT_D;                    // 1

  // norms, column sums of x, colsum(M)
  uot_rownorm2<<<16, 256, 0, stream>>>(x, xxv);
  uot_rownorm2<<<16, 256, 0, stream>>>(y, yyv);
  uot_xsum1<<<32, 256, 0, stream>>>(x, xpart);
  uot_xsum2<<<1, 256, 0, stream>>>(xpart, xsum);
  uot_vecsum<<<1, 256, 0, stream>>>(xxv, UOT_M, sxx);
  uot_colm<<<16, 256, 0, stream>>>(y, xsum, yyv, sxx, colM);

  // WMMA GEMM + K/G1 epilogue
  uot_gemm_k<<<dim3(UOT_N / 64, UOT_M / 128), 256, 0, stream>>>(x, y, xxv, yyv, Kmat, G);

  // R1, C1
  uot_rowsum<<<UOT_M, 256, 0, stream>>>(G, R);
  uot_colsum1<<<dim3(16, 8), 256, 0, stream>>>(G, Cpart);
  uot_colsum2<<<16, 256, 0, stream>>>(Cpart, C);

  // 29 remaining MM iterations (G1 -> G30); update fuses the new row sums
  for (int it = 0; it < 29; ++it) {
    uot_update<<<UOT_M, 256, 0, stream>>>(Kmat, G, R, C);
    uot_colsum1<<<dim3(16, 8), 256, 0, stream>>>(G, Cpart);
    uot_colsum2<<<16, 256, 0, stream>>>(Cpart, C);
  }

  // after the loop C == colsum(G30): final reduction
  uot_final<<<1, 256, 0, stream>>>(C, colM, out);
}